// SimpleAttentionLayer_11003706212957
// MI455X (gfx1250) — compile-verified
//
#include <hip/hip_runtime.h>

// ---------------------------------------------------------------------------
// SimpleAttentionLayer decode step for MI455X (gfx1250, wave32, WMMA).
// All three stages are HBM-bandwidth-bound at fp32 (~400MB total traffic,
// ~17us at 23.3TB/s), so we keep fp32 end-to-end and use V_WMMA_F32_16X16X4_F32
// for every matmul. Attention exploits PAGE_SIZE==16 == WMMA tile width:
// one page-table lookup per score tile; KV is read once per head-workgroup.
// The P*V phase streams V pages into LDS with gfx1250 async-to-LDS copies
// (double-buffered, s_wait_asynccnt-pipelined); GEMMs prefetch the weight
// stream with global_prefetch_b8 and use stepped pointers (no per-iter muls).
// ---------------------------------------------------------------------------

typedef float v2f __attribute__((ext_vector_type(2)));
typedef float v4f __attribute__((ext_vector_type(4)));
typedef float v8f __attribute__((ext_vector_type(8)));
typedef int   v4i __attribute__((ext_vector_type(4)));

typedef v4i __attribute__((address_space(1)))* gv4i_p;  // global int4*
typedef v4i __attribute__((address_space(3)))* lv4i_p;  // LDS int4*

#define NUM_HEADS 32
#define HEAD_DIM  128
#define HIDDEN    4096
#define QKV_N     (3 * HIDDEN)      // 12288
#define PAGE_SIZE 16
#define CACHE_POS 4096
#define NPAGES    (CACHE_POS / PAGE_SIZE)   // 256
#define BATCH     32
#define PSTR      4112               // padded LDS row stride for P (floats)
#define SM_SCALE  0.08838834764831845f   // 1/sqrt(128)

#if defined(__has_builtin)
#if __has_builtin(__builtin_amdgcn_global_load_async_to_lds_b128) && \
    __has_builtin(__builtin_amdgcn_s_wait_asynccnt)
#define USE_ASYNC_LDS 1
#endif
#endif

#ifndef USE_ASYNC_LDS
#define USE_ASYNC_LDS 0
#endif

#if USE_ASYNC_LDS
#define ASYNC_WAIT(n) __builtin_amdgcn_s_wait_asynccnt(n)
#else
#define ASYNC_WAIT(n) ((void)0)
#endif

__device__ __forceinline__ v8f wmma_f32(v2f a, v2f b, v8f c) {
  // 8 args: (neg_a, A, neg_b, B, c_mod, C, reuse_a, reuse_b)
  return __builtin_amdgcn_wmma_f32_16x16x4_f32(false, a, false, b, (short)0, c,
                                               false, false);
}

// Stage one V page tile [16 pos x 128 dims] (rows strided 4096 floats in
// global) into LDS. Async path: one b128 per lane, 512B per row, 2 rows/wave.
__device__ __forceinline__ void stage_vtile(float* dst, const float* src,
                                            int tid) {
  const int lane = tid & 31;
  const int w = tid >> 5;
#pragma unroll
  for (int p = 0; p < 2; ++p) {
    const int row = w + p * 8;
    const float* g = src + (size_t)row * (NUM_HEADS * HEAD_DIM) + lane * 4;
    float* l = dst + row * HEAD_DIM + lane * 4;
#if USE_ASYNC_LDS
    __builtin_amdgcn_global_load_async_to_lds_b128((gv4i_p)(float*)g,
                                                   (lv4i_p)l, 0, 0);
#else
    *(v4f*)l = *(const v4f*)g;
#endif
  }
}

// ---------------------------------------------------------------------------
// Generic C[M=32, N] = A[32, K] * B[K, N] + bias, one 16x16 tile per wave.
// A-frag (16x4 f32): lane r=lane&15 holds row r; lanes>=16 take K+2,K+3.
// B-frag (4x16 f32): lane n=lane&15 holds col n; lanes>=16 take K+2,K+3.
// B pointer steps by 4*N per k-step so addresses fold into immediate offsets.
// ---------------------------------------------------------------------------
__global__ void __launch_bounds__(256)
gemm_bias_wmma(const float* __restrict__ A, const float* __restrict__ B,
               const float* __restrict__ bias, float* __restrict__ C,
               int K, int N) {
  const int lane = threadIdx.x & 31;
  const int wave = blockIdx.x * (blockDim.x >> 5) + (threadIdx.x >> 5);
  const int mt = wave & 1;          // M=32 -> 2 tiles of 16
  const int nt = wave >> 1;
  if (nt * 16 >= N) return;

  const int r    = lane & 15;
  const int hi   = lane >> 4;       // 0: K+0/1, 1: K+2/3
  const int koff = hi * 2;

  const float* ap = A + (size_t)(mt * 16 + r) * K + koff;
  const float* bp = B + nt * 16 + r + (size_t)koff * N;
  const size_t bstep = (size_t)4 * N;   // floats advanced per k-step

  v8f acc = {};
  for (int k0 = 0; k0 < K; k0 += 64) {
    if (k0 + 64 < K) {  // prefetch next B slab of the streamed weight matrix
      __builtin_prefetch(bp + (size_t)64 * N, 0, 1);
      __builtin_prefetch(bp + (size_t)96 * N, 0, 1);
    }
#pragma unroll
    for (int u = 0; u < 16; ++u) {
      v2f a = *(const v2f*)ap;
      v2f b;
      b.x = bp[0];
      b.y = bp[(size_t)N];
      acc = wmma_f32(a, b, acc);
      ap += 4;
      bp += bstep;
    }
  }

  // D layout: VGPR v -> row = v + 8*hi, col = lane&15
  const int col = nt * 16 + r;
  const float bv = bias[col];
#pragma unroll
  for (int v = 0; v < 8; ++v) {
    const int row = mt * 16 + (hi << 3) + v;
    C[(size_t)row * N + col] = acc[v] + bv;
  }
}

// ---------------------------------------------------------------------------
// Attention: grid (2 batch halves, 32 heads), 256 threads = 8 waves.
// LDS: Q tile [16x128], P scores/probs [16 x PSTR], reductions, 2x V tile.
// ---------------------------------------------------------------------------
extern __shared__ float smem[];

__global__ void __launch_bounds__(256)
attn_paged_wmma(const float* __restrict__ qkv,
                const float* __restrict__ k_pages,
                const float* __restrict__ v_pages,
                const long long* __restrict__ page_table,
                float* __restrict__ attn_out) {
  const int half = blockIdx.x;          // batch half: rows 16*half .. +15
  const int h    = blockIdx.y;          // head
  const int tid  = threadIdx.x;
  const int lane = tid & 31;
  const int wave = tid >> 5;            // 0..7
  const int r    = lane & 15;
  const int hi   = lane >> 4;
  const int koff = hi * 2;

  float* qs     = smem;                       // 2048 floats
  float* P      = smem + 16 * HEAD_DIM;       // 16*PSTR
  float* red    = P + 16 * PSTR;              // 256
  float* rowred = red + 256;                  // 16
  float* vtbuf  = rowred + 16;                // 2 x 2048 (16B aligned)

  // Stage Q for this (half, head) into LDS.
  for (int i = tid; i < 16 * HEAD_DIM; i += 256) {
    const int rr = i >> 7, d = i & 127;
    qs[i] = qkv[(size_t)(half * 16 + rr) * QKV_N + h * HEAD_DIM + d];
  }
  __syncthreads();

  // ---- Scores: P[16, t] = (Q * K^T) * scale. One WMMA N-tile == one page.
  for (int tile = wave; tile < NPAGES; tile += 8) {
    const long long phys = page_table[tile];
    // K row for position (tile*16 + r): k_pages[phys][r][h][:]
    const float* kb =
        k_pages + (((size_t)phys * PAGE_SIZE + r) * NUM_HEADS + h) * HEAD_DIM;
    const float* qp = qs + r * HEAD_DIM;
    v8f acc = {};
#pragma unroll 8
    for (int k = 0; k < HEAD_DIM; k += 4) {
      const int kk = k + koff;
      v2f a = *(const v2f*)(qp + kk);                  // ds_load_b64
      v2f b = *(const v2f*)(kb + kk);                  // global b64
      acc = wmma_f32(a, b, acc);
    }
#pragma unroll
    for (int v = 0; v < 8; ++v) {
      const int row = (hi << 3) + v;
      P[row * PSTR + tile * PAGE_SIZE + r] = acc[v] * SM_SCALE;
    }
  }

  // ---- Tail column t = CACHE_POS: current-token K differs per batch row.
  if (tid < 16) {
    const int b = half * 16 + tid;
    const float* kc = qkv + (size_t)b * QKV_N + HIDDEN + h * HEAD_DIM;
    float s = 0.f;
    for (int d = 0; d < HEAD_DIM; ++d) s += qs[tid * HEAD_DIM + d] * kc[d];
    P[tid * PSTR + CACHE_POS] = s * SM_SCALE;
  }
  __syncthreads();

  // ---- Softmax over 4097 columns: 16 threads per row.
  const int srow = tid >> 4;
  const int sj   = tid & 15;
  float m = -1e30f;
  for (int c = sj; c < CACHE_POS + 1; c += 16)
    m = fmaxf(m, P[srow * PSTR + c]);
  red[tid] = m;
  __syncthreads();
  if (sj == 0) {
    float mm = red[tid];
    for (int j = 1; j < 16; ++j) mm = fmaxf(mm, red[tid + j]);
    rowred[srow] = mm;
  }
  __syncthreads();
  const float rmax = rowred[srow];
  float s = 0.f;
  for (int c = sj; c < CACHE_POS + 1; c += 16) {
    const float e = __expf(P[srow * PSTR + c] - rmax);
    P[srow * PSTR + c] = e;
    s += e;
  }
  red[tid] = s;
  __syncthreads();
  if (sj == 0) {
    float ss = 0.f;
    for (int j = 0; j < 16; ++j) ss += red[tid + j];
    rowred[srow] = ss;
  }
  __syncthreads();
  const float inv = 1.0f / rowred[srow];
  for (int c = sj; c < CACHE_POS + 1; c += 16) P[srow * PSTR + c] *= inv;
  __syncthreads();

  // ---- Output: out[16,128] = P[16,4096] * V + tail.
  // All 8 waves share one V page tile per step: stage to LDS with async
  // copies, double-buffered so page pg+1 streams while pg feeds WMMA.
  const int ncol = wave * 16;
  const float* prow = P + r * PSTR;     // A row (from LDS)
  float* vt0 = vtbuf;
  float* vt1 = vtbuf + PAGE_SIZE * HEAD_DIM;
  v8f acc = {};

  auto vpage_base = [&](int pg) {
    const long long phys = page_table[pg];
    return v_pages + (size_t)phys * (PAGE_SIZE * NUM_HEADS * HEAD_DIM) +
           h * HEAD_DIM;
  };
  auto do_page = [&](int pg, const float* vt) {
    const float* vb = vt + ncol + r;
    const float* pp = prow + pg * PAGE_SIZE;
#pragma unroll
    for (int sK = 0; sK < PAGE_SIZE; sK += 4) {
      const int off = sK + koff;
      v2f a = *(const v2f*)(pp + off);                 // ds_load_b64
      v2f b;
      b.x = vb[off * HEAD_DIM];                        // ds_load_b32
      b.y = vb[(off + 1) * HEAD_DIM];
      acc = wmma_f32(a, b, acc);
    }
  };

  stage_vtile(vt0, vpage_base(0), tid);                // prologue
  for (int pg = 0; pg < NPAGES - 1; ++pg) {
    float* cur = (pg & 1) ? vt1 : vt0;
    float* nxt = (pg & 1) ? vt0 : vt1;
    stage_vtile(nxt, vpage_base(pg + 1), tid);
    ASYNC_WAIT(2);            // page pg's 2 ops done; pg+1's still in flight
    __syncthreads();
    do_page(pg, cur);
    __syncthreads();          // protect cur before it is re-staged
  }
  ASYNC_WAIT(0);
  __syncthreads();
  do_page(NPAGES - 1, ((NPAGES - 1) & 1) ? vt1 : vt0);

  // Tail: += P[row][4096] * v_cur(batch row), then store.
#pragma unroll
  for (int v = 0; v < 8; ++v) {
    const int row = (hi << 3) + v;
    const int b   = half * 16 + row;
    const float p = P[row * PSTR + CACHE_POS];
    const float vv =
        qkv[(size_t)b * QKV_N + 2 * HIDDEN + h * HEAD_DIM + ncol + r];
    acc[v] += p * vv;
    attn_out[((size_t)b * NUM_HEADS + h) * HEAD_DIM + ncol + r] = acc[v];
  }
}

// ---------------------------------------------------------------------------
extern "C" void kernel_launch(void* const* d_in, const int* in_sizes, int n_in,
                              void* d_out, int out_size, void* d_ws,
                              size_t ws_size, hipStream_t stream) {
  const float*     x          = (const float*)d_in[0];
  const float*     w_qkv      = (const float*)d_in[1];
  const float*     b_qkv      = (const float*)d_in[2];
  const float*     w_proj     = (const float*)d_in[3];
  const float*     b_proj     = (const float*)d_in[4];
  const float*     k_pages    = (const float*)d_in[5];
  const float*     v_pages    = (const float*)d_in[6];
  const long long* page_table = (const long long*)d_in[7];
  float* out = (float*)d_out;

  float* qkv      = (float*)d_ws;                    // [32, 12288]
  float* attn_out = qkv + (size_t)BATCH * QKV_N;     // [32, 32, 128] == [32,4096]

  // 1) QKV projection: 2*768 = 1536 wave-tiles -> 192 blocks x 8 waves.
  gemm_bias_wmma<<<192, 256, 0, stream>>>(x, w_qkv, b_qkv, qkv, HIDDEN, QKV_N);

  // 2) Paged attention: one head x batch-half per workgroup.
  const size_t lds_bytes =
      (size_t)(16 * HEAD_DIM + 16 * PSTR + 256 + 16 + 2 * PAGE_SIZE * HEAD_DIM) *
      sizeof(float);
  dim3 g(2, NUM_HEADS);
  attn_paged_wmma<<<g, 256, lds_bytes, stream>>>(qkv, k_pages, v_pages,
                                                 page_table, attn_out);

  // 3) Output projection: 2*256 = 512 wave-tiles -> 64 blocks x 8 waves.
  gemm_bias_wmma<<<64, 256, 0, stream>>>(attn_out, w_proj, b_proj, out,
                                         HIDDEN, HIDDEN);
}